// GAT2_65231963291900
// MI455X (gfx1250) — compile-verified
//
#include <hip/hip_runtime.h>

typedef __attribute__((ext_vector_type(2))) float v2f;
typedef __attribute__((ext_vector_type(8))) float v8f;

#define NEG_SLOPE 0.2f

static __device__ __forceinline__ void atomicMaxF(float* addr, float value) {
  // Order-preserving int punning: works for mixed signs with -inf init.
  if (value >= 0.0f) atomicMax((int*)addr, __float_as_int(value));
  else               atomicMin((unsigned int*)addr, __float_as_uint(value));
}

// ------------------------------ fills ------------------------------
__global__ void fill_val_kernel(float* __restrict__ p, float v, int n) {
  int i = blockIdx.x * blockDim.x + threadIdx.x;
  if (i < n) p[i] = v;
}

__global__ void fill_bias_kernel(float* __restrict__ p, const float* __restrict__ b,
                                 int n, int dim) {
  int i = blockIdx.x * blockDim.x + threadIdx.x;
  if (i < n) p[i] = b[i % dim];
}

// --------------------- GEMM: H = X(128) @ W(128xNCOL) ---------------------
// One 16-row M tile per block; each wave owns one 16-col N tile.
// A fragment (fp32 16x4): lanes 0-15 = M rows, VGPR0 holds K=+0 (lo half)/K=+2 (hi half),
// VGPR1 holds K=+1 / K=+3.  D: VGPR r -> M = r (lanes 0-15) / r+8 (lanes 16-31), N = lane&15.
template <int NCOL>
__global__ void gemm_wmma_kernel(const float* __restrict__ X,
                                 const float* __restrict__ W,
                                 float* __restrict__ Hout, int nrows) {
  __shared__ float xs[16][132];  // pad to 132 -> conflict-free column reads
  const int m0 = blockIdx.x * 16;
  const int t = threadIdx.x;
  for (int i = t; i < 16 * 128; i += blockDim.x) {
    int r = i >> 7, c = i & 127;
    int row = m0 + r;
    xs[r][c] = (row < nrows) ? X[(size_t)row * 128 + c] : 0.0f;
  }
  __syncthreads();

  const int wave = t >> 5;
  const int lane = t & 31;
  const int kh   = lane >> 4;   // half-wave selector
  const int l15  = lane & 15;
  const int n0   = wave * 16;

  v8f acc = {0.f, 0.f, 0.f, 0.f, 0.f, 0.f, 0.f, 0.f};
#pragma unroll
  for (int k0 = 0; k0 < 128; k0 += 4) {
    v2f a, b;
    a.x = xs[l15][k0 + 2 * kh];
    a.y = xs[l15][k0 + 2 * kh + 1];
    b.x = W[(k0 + 2 * kh) * NCOL + n0 + l15];
    b.y = W[(k0 + 2 * kh + 1) * NCOL + n0 + l15];
    acc = __builtin_amdgcn_wmma_f32_16x16x4_f32(false, a, false, b,
                                                (short)0, acc, false, false);
  }
#pragma unroll
  for (int r = 0; r < 8; ++r) {
    int row = m0 + r + 8 * kh;
    if (row < nrows) Hout[(size_t)row * NCOL + n0 + l15] = acc[r];
  }
}

// -------------------- per-node attention coefficients --------------------
// asrc[n,h] = sum_c H[n,h,c]*a_src[h,c] ; adst likewise.
__global__ void alpha_kernel(const float* __restrict__ Hf,
                             const float* __restrict__ a_src,
                             const float* __restrict__ a_dst,
                             float* __restrict__ asrc, float* __restrict__ adst,
                             int n, int Hh, int C) {
  int t = blockIdx.x * blockDim.x + threadIdx.x;
  if (t >= n * Hh) return;
  int node = t / Hh, head = t % Hh;
  const float* hp = Hf + (size_t)node * Hh * C + (size_t)head * C;
  const float* as = a_src + head * C;
  const float* ad = a_dst + head * C;
  float s0 = 0.f, s1 = 0.f;
  for (int j = 0; j < C; ++j) {
    float v = hp[j];
    s0 += v * as[j];
    s1 += v * ad[j];
  }
  asrc[t] = s0;
  adst[t] = s1;
}

static __device__ __forceinline__ void edge_sd(const int* __restrict__ ei, int e, int E,
                                               int& s, int& d) {
  if (e < E) { s = ei[e]; d = ei[E + e]; }      // edge_index row-major (2,E)
  else       { s = e - E; d = e - E; }          // synthesized self-loop
}

// pass 1: segment max of leaky_relu(asrc[s]+adst[d]) by destination
__global__ void edge_max_kernel(const int* __restrict__ ei, int E, int total,
                                const float* __restrict__ asrc,
                                const float* __restrict__ adst,
                                float* __restrict__ emax, int Hh) {
  int e = blockIdx.x * blockDim.x + threadIdx.x;
  if (e >= total) return;
  int s, d; edge_sd(ei, e, E, s, d);
  for (int h = 0; h < Hh; ++h) {
    float v = asrc[s * Hh + h] + adst[d * Hh + h];
    v = (v > 0.f) ? v : NEG_SLOPE * v;
    atomicMaxF(&emax[d * Hh + h], v);
  }
}

// pass 2: segment sum of exp(e - emax[d])
__global__ void edge_den_kernel(const int* __restrict__ ei, int E, int total,
                                const float* __restrict__ asrc,
                                const float* __restrict__ adst,
                                const float* __restrict__ emax,
                                float* __restrict__ den, int Hh) {
  int e = blockIdx.x * blockDim.x + threadIdx.x;
  if (e >= total) return;
  int s, d; edge_sd(ei, e, E, s, d);
  for (int h = 0; h < Hh; ++h) {
    float v = asrc[s * Hh + h] + adst[d * Hh + h];
    v = (v > 0.f) ? v : NEG_SLOPE * v;
    atomicAdd(&den[d * Hh + h], expf(v - emax[d * Hh + h]));
  }
}

// pass 3: wave per edge; lanes cover channels; atomic scatter-add of messages.
template <int OUT, int C, int Hh>
__global__ void edge_scatter_kernel(const int* __restrict__ ei, int E, int total,
                                    const float* __restrict__ asrc,
                                    const float* __restrict__ adst,
                                    const float* __restrict__ emax,
                                    const float* __restrict__ den,
                                    const float* __restrict__ feat,
                                    float* __restrict__ out) {
  int wid  = (blockIdx.x * blockDim.x + threadIdx.x) >> 5;
  int lane = threadIdx.x & 31;
  if (wid >= total) return;
  int s, d; edge_sd(ei, wid, E, s, d);
  float alpha[Hh];
#pragma unroll
  for (int h = 0; h < Hh; ++h) {
    float v = asrc[s * Hh + h] + adst[d * Hh + h];
    v = (v > 0.f) ? v : NEG_SLOPE * v;
    alpha[h] = expf(v - emax[d * Hh + h]) / den[d * Hh + h];
  }
#pragma unroll
  for (int q = 0; q < OUT / 32; ++q) {
    int ch = q * 32 + lane;
    int h  = ch / C;
    float val = feat[(size_t)s * OUT + ch] * alpha[h];
    atomicAdd(&out[(size_t)d * OUT + ch], val);
  }
}

// bias + ELU, in place
__global__ void bias_elu_kernel(float* __restrict__ buf, const float* __restrict__ b,
                                int n, int dim) {
  int i = blockIdx.x * blockDim.x + threadIdx.x;
  if (i >= n) return;
  float v = buf[i] + b[i % dim];
  buf[i] = (v > 0.f) ? v : (expf(v) - 1.0f);
}

static inline int cdiv(int a, int b) { return (a + b - 1) / b; }

extern "C" void kernel_launch(void* const* d_in, const int* in_sizes, int n_in,
                              void* d_out, int out_size, void* d_ws, size_t ws_size,
                              hipStream_t stream) {
  const float* x      = (const float*)d_in[0];
  const int*   ei     = (const int*)  d_in[1];
  const float* W1     = (const float*)d_in[2];
  const float* a_src1 = (const float*)d_in[3];
  const float* a_dst1 = (const float*)d_in[4];
  const float* b1     = (const float*)d_in[5];
  const float* W2     = (const float*)d_in[6];
  const float* a_src2 = (const float*)d_in[7];
  const float* a_dst2 = (const float*)d_in[8];
  const float* b2     = (const float*)d_in[9];
  float* out = (float*)d_out;

  const int N  = in_sizes[0] / 128;   // 100000
  const int E  = in_sizes[1] / 2;     // 1600000
  const int TE = E + N;               // with self-loops

  // workspace layout (floats)
  float* w     = (float*)d_ws;
  float* h1    = w;                       // N*128  (reused as h2: N*64)
  float* agg1  = h1   + (size_t)N * 128;  // N*128  (becomes post-ELU features)
  float* asrc1 = agg1 + (size_t)N * 128;  // N*4
  float* adst1 = asrc1 + (size_t)N * 4;
  float* emax1 = adst1 + (size_t)N * 4;
  float* den1  = emax1 + (size_t)N * 4;
  float* asrc2 = den1  + (size_t)N * 4;   // N*1
  float* adst2 = asrc2 + (size_t)N;
  float* emax2 = adst2 + (size_t)N;
  float* den2  = emax2 + (size_t)N;
  float* h2    = h1;                      // reuse after layer-1 scatter

  const float ninf = -__builtin_inff();

  // ---------------- layer 1 ----------------
  fill_val_kernel<<<cdiv(N * 4, 256), 256, 0, stream>>>(emax1, ninf, N * 4);
  fill_val_kernel<<<cdiv(N * 4, 256), 256, 0, stream>>>(den1, 0.0f, N * 4);
  fill_val_kernel<<<cdiv(N * 128, 256), 256, 0, stream>>>(agg1, 0.0f, N * 128);

  gemm_wmma_kernel<128><<<cdiv(N, 16), 256, 0, stream>>>(x, W1, h1, N);

  alpha_kernel<<<cdiv(N * 4, 256), 256, 0, stream>>>(h1, a_src1, a_dst1,
                                                     asrc1, adst1, N, 4, 32);

  edge_max_kernel<<<cdiv(TE, 256), 256, 0, stream>>>(ei, E, TE, asrc1, adst1, emax1, 4);
  edge_den_kernel<<<cdiv(TE, 256), 256, 0, stream>>>(ei, E, TE, asrc1, adst1, emax1, den1, 4);
  edge_scatter_kernel<128, 32, 4><<<cdiv(TE, 8), 256, 0, stream>>>(
      ei, E, TE, asrc1, adst1, emax1, den1, h1, agg1);

  bias_elu_kernel<<<cdiv(N * 128, 256), 256, 0, stream>>>(agg1, b1, N * 128, 128);

  // ---------------- layer 2 ----------------
  fill_val_kernel<<<cdiv(N, 256), 256, 0, stream>>>(emax2, ninf, N);
  fill_val_kernel<<<cdiv(N, 256), 256, 0, stream>>>(den2, 0.0f, N);
  fill_bias_kernel<<<cdiv(N * 64, 256), 256, 0, stream>>>(out, b2, N * 64, 64);

  gemm_wmma_kernel<64><<<cdiv(N, 16), 128, 0, stream>>>(agg1, W2, h2, N);

  alpha_kernel<<<cdiv(N, 256), 256, 0, stream>>>(h2, a_src2, a_dst2,
                                                 asrc2, adst2, N, 1, 64);

  edge_max_kernel<<<cdiv(TE, 256), 256, 0, stream>>>(ei, E, TE, asrc2, adst2, emax2, 1);
  edge_den_kernel<<<cdiv(TE, 256), 256, 0, stream>>>(ei, E, TE, asrc2, adst2, emax2, den2, 1);
  edge_scatter_kernel<64, 64, 1><<<cdiv(TE, 8), 256, 0, stream>>>(
      ei, E, TE, asrc2, adst2, emax2, den2, h2, out);
}